// AdaptiveEmbedding_53197464928440
// MI455X (gfx1250) — compile-verified
//
#include <hip/hip_runtime.h>

typedef __attribute__((ext_vector_type(2))) float v2f;
typedef __attribute__((ext_vector_type(8))) float v8f;

#define CUT0 20000
#define CUT1 60000
#define D_MODEL 1024
#define K1 256
#define K2 64
#define KTOT 320            // fused K = 256 (proj1) + 64 (proj2)
#define LDS_STRIDE 322      // even (b64 aligned) and %64==2 (low bank conflicts)

__global__ __launch_bounds__(128)
void adaptive_emb_wmma(const int* __restrict__ ids,
                       const float* __restrict__ emb0,
                       const float* __restrict__ emb1,
                       const float* __restrict__ emb2,
                       const float* __restrict__ proj1,
                       const float* __restrict__ proj2,
                       float* __restrict__ out,
                       int nTok)
{
    __shared__ float A[16 * LDS_STRIDE];
    __shared__ int sCl[16];     // cluster id per row
    __shared__ int sLoc[16];    // local table row per row
    __shared__ int sLoc0[16];   // emb0 row (>=0) for cluster-0 tokens, else -1

    const int tid  = threadIdx.x;
    const int lane = tid & 31;
    const int wave = tid >> 5;
    const int tileBase = blockIdx.x * 16;

    // ---- per-token routing metadata ----
    if (tid < 16) {
        int t  = tileBase + tid;
        int id = (t < nTok) ? ids[t] : 0;
        int cl, loc;
        if (id < CUT0)      { cl = 0; loc = 0; }
        else if (id < CUT1) { cl = 1; loc = id - CUT0; }
        else                { cl = 2; loc = id - CUT1; }
        sCl[tid]   = cl;
        sLoc[tid]  = loc;
        sLoc0[tid] = (cl == 0 && id > 0) ? id : -1;
    }
    __syncthreads();

    // ---- build 16 x 320 fused A-tile in LDS ----
    // cluster1 row -> emb1 row in cols [0,256); cluster2 row -> emb2 row in [256,320);
    // cluster0 row -> zeros (direct emb0 add happens at store time)
    for (int idx = tid; idx < 16 * KTOT; idx += 128) {
        int r  = idx / KTOT;
        int c  = idx - r * KTOT;
        int cl = sCl[r];
        float v = 0.0f;
        if (cl == 1 && c < K1)  v = emb1[sLoc[r] * K1 + c];
        if (cl == 2 && c >= K1) v = emb2[sLoc[r] * K2 + (c - K1)];
        A[r * LDS_STRIDE + c] = v;
    }
    __syncthreads();

    const int laneN = lane & 15;
    const int kHalf = (lane >> 4) << 1;   // 0 for lanes 0-15, 2 for lanes 16-31
    const int mOff  = (lane >> 4) << 3;   // C/D rows 0-7 vs 8-15

    // ---- each wave owns 256 output columns: 4 groups of 4 WMMA tiles ----
    for (int g = 0; g < 4; ++g) {
        const int n0 = wave * 256 + g * 64;
        v8f acc[4] = {};

        // K in [0,256): B = proj1^T (proj1 is [1024 x 256] row-major -> k contiguous)
        #pragma unroll 4
        for (int k0 = 0; k0 < K1; k0 += 4) {
            const int kb = k0 + kHalf;
            v2f a = *(const v2f*)&A[laneN * LDS_STRIDE + kb];
            #pragma unroll
            for (int j = 0; j < 4; ++j) {
                v2f b = *(const v2f*)(proj1 + (n0 + j * 16 + laneN) * K1 + kb);
                acc[j] = __builtin_amdgcn_wmma_f32_16x16x4_f32(
                    false, a, false, b, (short)0, acc[j], false, false);
            }
        }
        // K in [256,320): B = proj2^T (proj2 is [1024 x 64] row-major)
        #pragma unroll 4
        for (int k0 = K1; k0 < KTOT; k0 += 4) {
            const int kb = k0 + kHalf;
            v2f a = *(const v2f*)&A[laneN * LDS_STRIDE + kb];
            #pragma unroll
            for (int j = 0; j < 4; ++j) {
                v2f b = *(const v2f*)(proj2 + (n0 + j * 16 + laneN) * K2 + (kb - K1));
                acc[j] = __builtin_amdgcn_wmma_f32_16x16x4_f32(
                    false, a, false, b, (short)0, acc[j], false, false);
            }
        }

        // ---- store D * sqrt(1024), adding direct emb0 rows for cluster-0 tokens ----
        #pragma unroll
        for (int j = 0; j < 4; ++j) {
            const int col = n0 + j * 16 + laneN;
            #pragma unroll
            for (int r = 0; r < 8; ++r) {
                int m = r + mOff;
                int t = tileBase + m;
                float v = acc[j][r];
                int l0 = sLoc0[m];
                if (l0 >= 0) v += emb0[l0 * D_MODEL + col];
                if (t < nTok) out[t * D_MODEL + col] = v * 32.0f;
            }
        }
    }
}

extern "C" void kernel_launch(void* const* d_in, const int* in_sizes, int n_in,
                              void* d_out, int out_size, void* d_ws, size_t ws_size,
                              hipStream_t stream) {
    const int*   ids   = (const int*)d_in[0];
    const float* emb0  = (const float*)d_in[1];
    const float* emb1  = (const float*)d_in[2];
    const float* emb2  = (const float*)d_in[3];
    const float* proj1 = (const float*)d_in[4];
    const float* proj2 = (const float*)d_in[5];
    float* out = (float*)d_out;

    const int nTok   = in_sizes[0];            // B * S
    const int nTiles = (nTok + 15) / 16;

    adaptive_emb_wmma<<<nTiles, 128, 0, stream>>>(
        ids, emb0, emb1, emb2, proj1, proj2, out, nTok);
}